// ConditionalNF_22832046146152
// MI455X (gfx1250) — compile-verified
//
#include <hip/hip_runtime.h>
#include <hip/hip_bf16.h>
#include <math.h>

// Problem dims
#define BS   256
#define NS   512
#define D_   64
#define L_   16
#define XDIM 512
#define H_   512
#define RAWN 2192   // 2*64 + 2*16*64 + 16

typedef __attribute__((ext_vector_type(2))) float v2f;
typedef __attribute__((ext_vector_type(8))) float v8f;

__device__ __forceinline__ float softplusf(float x) {
    return x > 20.f ? x : log1pf(expf(x));
}

__device__ __forceinline__ float wave_reduce_add(float v) {
    #pragma unroll
    for (int m = 16; m >= 1; m >>= 1) v += __shfl_xor(v, m, 32);
    return v;
}

// ---------------------------------------------------------------------------
// C[M x N] = act(A[M x K] @ W[K x N] + bias[N]),  act: 1 = ELU, 0 = identity
// One wave (32 lanes) computes one 16x16 tile with V_WMMA_F32_16X16X4_F32.
// block = (32,4): 4 waves -> 4 N-tiles; grid = (M/16, ceil(Ntiles/4)).
// At these sizes (512..2192 tiles) one-tile-per-wave maximizes wave count.
// ---------------------------------------------------------------------------
__global__ void gemm_wmma_f32_kernel(const float* __restrict__ A,
                                     const float* __restrict__ W,
                                     const float* __restrict__ bias,
                                     float* __restrict__ C,
                                     int K, int N, int Ntiles, int act) {
    const int tileM = blockIdx.x;
    const int tileN = blockIdx.y * 4 + threadIdx.y;
    if (tileN >= Ntiles) return;          // whole wave exits -> EXEC stays all-1s

    const int lane = threadIdx.x;          // 0..31
    const int half = lane >> 4;            // K-pair select: 0 -> K{0,1}, 1 -> K{2,3}
    const int m16  = lane & 15;

    // A fragment source: A[tileM*16 + m16][k + 2*half .. +1]  (float2)
    const float* arow = A + (size_t)(tileM * 16 + m16) * K + 2 * half;
    // B fragment source: W[k + 2*half .. +1][tileN*16 + m16]  (two strided b32)
    const int col = tileN * 16 + m16;
    const float* bcol = W + (size_t)(2 * half) * N + col;

    v8f acc = {0.f, 0.f, 0.f, 0.f, 0.f, 0.f, 0.f, 0.f};

    #pragma unroll 8
    for (int k = 0; k < K; k += 4) {
        float2 av = *(const float2*)(arow + k);      // 8B aligned b64 load
        v2f a; a.x = av.x; a.y = av.y;
        v2f b; b.x = bcol[(size_t)k * N];
               b.y = bcol[(size_t)(k + 1) * N];
        acc = __builtin_amdgcn_wmma_f32_16x16x4_f32(
                  false, a, false, b, (short)0, acc, false, false);
    }

    const float bv = bias[col];
    #pragma unroll
    for (int r = 0; r < 8; r++) {
        // D layout: VGPR r -> M = r (lanes 0-15) / r+8 (lanes 16-31), N = lane%16
        const int row = tileM * 16 + half * 8 + r;
        float v = acc[r] + bv;
        if (act) v = v > 0.f ? v : expm1f(v);        // ELU
        C[(size_t)row * N + col] = v;
    }
}

// ---------------------------------------------------------------------------
// Per-(b,l) flow parameters (sample-independent -> hoisted out of the flow):
//   wdu = w.u ; m = -1 + softplus(wdu + 0.5413) ; u_hat = u + (m-wdu) w / |w|^2
//   wuh = w.u_hat
// One wave32 per (b,l), 2 dims per lane, shuffle reductions.
// ---------------------------------------------------------------------------
__global__ void prep_kernel(const float* __restrict__ raw,
                            float* __restrict__ u_hat,
                            float* __restrict__ wuh) {
    const int id = blockIdx.x;          // 0 .. BS*L_-1
    const int b = id >> 4, l = id & 15;
    const int lane = threadIdx.x;

    const float* wb = raw + (size_t)b * RAWN + 2 * D_ + l * D_;           // w[b,l,:]
    const float* ub = raw + (size_t)b * RAWN + 2 * D_ + L_ * D_ + l * D_; // u[b,l,:]

    const float w0 = wb[lane], w1 = wb[lane + 32];
    const float u0 = ub[lane], u1 = ub[lane + 32];

    const float wdu = wave_reduce_add(w0 * u0 + w1 * u1);
    const float w2  = wave_reduce_add(w0 * w0 + w1 * w1);

    const float m = softplusf(wdu + 0.5413f) - 1.0f;
    const float c = (m - wdu) / w2;
    const float uh0 = u0 + c * w0;
    const float uh1 = u1 + c * w1;

    float* uo = u_hat + ((size_t)b * L_ + l) * D_;
    uo[lane]      = uh0;
    uo[lane + 32] = uh1;

    const float s = wave_reduce_add(w0 * uh0 + w1 * uh1);
    if (lane == 0) wuh[b * L_ + l] = s;
}

// ---------------------------------------------------------------------------
// Per-b: sigma = softplus(raw[:,64:128]), sumlog = sum_d log(sigma)
// ---------------------------------------------------------------------------
__global__ void sigma_kernel(const float* __restrict__ raw,
                             float* __restrict__ sigma,
                             float* __restrict__ sumlog) {
    const int b = blockIdx.x;
    const int lane = threadIdx.x;
    const float p0 = raw[(size_t)b * RAWN + D_ + lane];
    const float p1 = raw[(size_t)b * RAWN + D_ + lane + 32];
    const float s0 = softplusf(p0);
    const float s1 = softplusf(p1);
    sigma[b * D_ + lane]      = s0;
    sigma[b * D_ + lane + 32] = s1;
    const float t = wave_reduce_add(logf(s0) + logf(s1));
    if (lane == 0) sumlog[b] = t;
}

// ---------------------------------------------------------------------------
// Flow: one block per batch element b, 512 threads = 512 samples.
// z[64] lives in VGPRs; per-layer params broadcast from LDS via ds_load_b128
// (uniform address -> conflict-free broadcast, 4 values per DS op).
// 16 sequential planar-flow steps; the z.w dot product uses 4 independent
// accumulators to break the FMA latency chain.
// ---------------------------------------------------------------------------
__global__ void __launch_bounds__(NS) flow_kernel(const float* __restrict__ raw,
                                                  const float* __restrict__ u_hat,
                                                  const float* __restrict__ wuh,
                                                  const float* __restrict__ sigma,
                                                  const float* __restrict__ sumlog,
                                                  const float* __restrict__ eps,
                                                  float* __restrict__ out) {
    __shared__ __align__(16) float s_w[L_ * D_];
    __shared__ __align__(16) float s_uh[L_ * D_];
    __shared__ __align__(16) float s_mu[D_];
    __shared__ __align__(16) float s_sig[D_];
    __shared__ float s_bl[L_];
    __shared__ float s_wuh[L_];
    __shared__ float s_slog;

    const int b = blockIdx.x;
    const int n = threadIdx.x;                 // sample index within batch elem
    const float* rb = raw + (size_t)b * RAWN;

    // stage per-b parameters
    for (int i = n; i < L_ * D_; i += NS) {
        s_w[i]  = rb[2 * D_ + i];              // w block is contiguous 1024 floats
        s_uh[i] = u_hat[(size_t)b * L_ * D_ + i];
    }
    if (n < D_) { s_mu[n] = rb[n]; s_sig[n] = sigma[b * D_ + n]; }
    if (n < L_) { s_bl[n] = rb[2 * D_ + 2 * L_ * D_ + n]; s_wuh[n] = wuh[b * L_ + n]; }
    if (n == 0) { s_slog = sumlog[b]; }
    __syncthreads();

    // z0 = mu + sigma * eps ; lp0 = -0.5*sum(eps^2) - sum(log sigma) - 32*ln(2pi)
    float z[D_];
    float e0 = 0.f, e1 = 0.f, e2 = 0.f, e3 = 0.f;
    const float4* ep = (const float4*)(eps + ((size_t)b * NS + n) * D_);
    #pragma unroll
    for (int d = 0; d < D_; d += 4) {
        const float4 e = ep[d >> 2];
        z[d + 0] = fmaf(s_sig[d + 0], e.x, s_mu[d + 0]); e0 = fmaf(e.x, e.x, e0);
        z[d + 1] = fmaf(s_sig[d + 1], e.y, s_mu[d + 1]); e1 = fmaf(e.y, e.y, e1);
        z[d + 2] = fmaf(s_sig[d + 2], e.z, s_mu[d + 2]); e2 = fmaf(e.z, e.z, e2);
        z[d + 3] = fmaf(s_sig[d + 3], e.w, s_mu[d + 3]); e3 = fmaf(e.w, e.w, e3);
    }
    float lp = -0.5f * ((e0 + e1) + (e2 + e3)) - s_slog - 58.8120661f; // 32*ln(2pi)

    #pragma unroll 1
    for (int l = 0; l < L_; l++) {
        const float4* wl4 = (const float4*)&s_w[l << 6];
        const float4* ul4 = (const float4*)&s_uh[l << 6];
        // dot(z, w) with 4 independent accumulators (breaks FMA latency chain)
        float p0 = 0.f, p1 = 0.f, p2 = 0.f, p3 = 0.f;
        #pragma unroll
        for (int q = 0; q < D_ / 4; q++) {
            const float4 wv = wl4[q];                  // ds_load_b128 broadcast
            p0 = fmaf(z[4 * q + 0], wv.x, p0);
            p1 = fmaf(z[4 * q + 1], wv.y, p1);
            p2 = fmaf(z[4 * q + 2], wv.z, p2);
            p3 = fmaf(z[4 * q + 3], wv.w, p3);
        }
        const float pre = s_bl[l] + ((p0 + p1) + (p2 + p3));
        const float t = tanhf(pre);
        #pragma unroll
        for (int q = 0; q < D_ / 4; q++) {
            const float4 uv = ul4[q];                  // ds_load_b128 broadcast
            z[4 * q + 0] = fmaf(uv.x, t, z[4 * q + 0]);
            z[4 * q + 1] = fmaf(uv.y, t, z[4 * q + 1]);
            z[4 * q + 2] = fmaf(uv.z, t, z[4 * q + 2]);
            z[4 * q + 3] = fmaf(uv.w, t, z[4 * q + 3]);
        }
        const float psi = (1.f - t * t) * s_wuh[l];
        lp -= logf(fabsf(1.f + psi));
    }

    float4* oz = (float4*)(out + ((size_t)b * NS + n) * D_);
    #pragma unroll
    for (int d = 0; d < D_; d += 4) {
        float4 v; v.x = z[d]; v.y = z[d + 1]; v.z = z[d + 2]; v.w = z[d + 3];
        oz[d >> 2] = v;
    }
    out[(size_t)BS * NS * D_ + (size_t)b * NS + n] = lp;
}

// ---------------------------------------------------------------------------
extern "C" void kernel_launch(void* const* d_in, const int* in_sizes, int n_in,
                              void* d_out, int out_size, void* d_ws, size_t ws_size,
                              hipStream_t stream) {
    const float* x   = (const float*)d_in[0];
    const float* eps = (const float*)d_in[1];
    const float* W1  = (const float*)d_in[2];
    const float* b1  = (const float*)d_in[3];
    const float* W2  = (const float*)d_in[4];
    const float* b2  = (const float*)d_in[5];
    const float* W3  = (const float*)d_in[6];
    const float* b3  = (const float*)d_in[7];

    float* ws = (float*)d_ws;
    float* h1     = ws;                         // 256*512   = 131072
    float* h2     = ws + 131072;                // 256*512   = 131072
    float* raw    = ws + 262144;                // 256*2192  = 561152
    float* uh     = ws + 823296;                // 256*16*64 = 262144
    float* wuhp   = ws + 1085440;               // 256*16    = 4096
    float* sig    = ws + 1089536;               // 256*64    = 16384
    float* slog   = ws + 1105920;               // 256       = 256

    const dim3 blk(32, 4);
    // h1 = elu(x @ W1 + b1)          M=256,K=512,N=512 (32 N-tiles)
    gemm_wmma_f32_kernel<<<dim3(16, 8), blk, 0, stream>>>(x,  W1, b1, h1, XDIM, H_, 32, 1);
    // h2 = elu(h1 @ W2 + b2)
    gemm_wmma_f32_kernel<<<dim3(16, 8), blk, 0, stream>>>(h1, W2, b2, h2, H_, H_, 32, 1);
    // raw = h2 @ W3 + b3             N=2192 (137 N-tiles)
    gemm_wmma_f32_kernel<<<dim3(16, 35), blk, 0, stream>>>(h2, W3, b3, raw, H_, RAWN, 137, 0);

    prep_kernel<<<BS * L_, 32, 0, stream>>>(raw, uh, wuhp);
    sigma_kernel<<<BS, 32, 0, stream>>>(raw, sig, slog);

    flow_kernel<<<BS, NS, 0, stream>>>(raw, uh, wuhp, sig, slog, eps, (float*)d_out);
}